// Caps_40424232190340
// MI455X (gfx1250) — compile-verified
//
#include <hip/hip_runtime.h>

typedef __attribute__((ext_vector_type(2))) float v2f;
typedef __attribute__((ext_vector_type(8))) float v8f;

#define B_SZ 64
#define I_SZ 1024
#define DIN  16
#define C_SZ 32
#define DOUT 16

// ---------------------------------------------------------------------------
// Kernel A: inputs_hat[b,i,c,e] = sum_d inputs[b,i,d] * W[i,c,d,e]
// Per-i GEMM: M=64 (batch), K=16 (DIN), N=512 (C*DOUT) using
// V_WMMA_F32_16X16X4_F32 (exact f32, 4 chained K=4 steps per tile).
// Block = one i, 256 threads = 8 wave32 waves. Wave w owns c in [4w,4w+4).
// ---------------------------------------------------------------------------
__global__ __launch_bounds__(256) void caps_hat_wmma(
    const float* __restrict__ in,   // (B, I, DIN)
    const float* __restrict__ W,    // (I, C, DIN, DOUT)
    float* __restrict__ hat)        // (B, I, C, DOUT)
{
  const int i      = blockIdx.x;
  const int tid    = threadIdx.x;
  const int wave   = tid >> 5;
  const int lane   = tid & 31;
  const int half   = lane >> 4;   // 0: lanes 0-15, 1: lanes 16-31
  const int lane16 = lane & 15;

  // Stage inputs[:, i, :] (64x16 f32) in LDS, stride 17 to avoid bank conflicts.
  __shared__ float Ash[64 * 17];
  for (int idx = tid; idx < 64 * 16; idx += 256) {
    int row = idx >> 4, d = idx & 15;
    Ash[row * 17 + d] = in[((size_t)row * I_SZ + i) * DIN + d];
  }
  __syncthreads();

  const float* Wi = W + (size_t)i * C_SZ * DIN * DOUT;

  for (int cl = 0; cl < 4; ++cl) {
    const int c = (wave << 2) + cl;

    // B fragments for this c, all 4 K-steps. B is 4x16 (KxN) per step:
    // lanes 0-15 hold K = ks*4+{0,1}, lanes 16-31 hold K = ks*4+{2,3}; N = lane16.
    v2f bf[4];
#pragma unroll
    for (int ks = 0; ks < 4; ++ks) {
      const int k = ks * 4 + 2 * half;
      const float* p = Wi + ((size_t)c * DIN + k) * DOUT + lane16;
      bf[ks].x = p[0];      // W[i, c, k,   lane16]
      bf[ks].y = p[DOUT];   // W[i, c, k+1, lane16]
    }

#pragma unroll
    for (int m = 0; m < 4; ++m) {
      v8f acc = {};  // C = 0
#pragma unroll
      for (int ks = 0; ks < 4; ++ks) {
        const int k   = ks * 4 + 2 * half;
        const int row = (m << 4) + lane16;  // A: M = row, lanes 0-15 K={k,k+1}, 16-31 K={k+2,k+3}
        v2f af;
        af.x = Ash[row * 17 + k];
        af.y = Ash[row * 17 + k + 1];
        acc = __builtin_amdgcn_wmma_f32_16x16x4_f32(
            /*neg_a=*/false, af, /*neg_b=*/false, bf[ks],
            /*c_mod=*/(short)0, acc, /*reuse_a=*/false, /*reuse_b=*/false);
      }
      // D layout: VGPR r -> M = m*16 + r + 8*half, N (=e) = lane16
#pragma unroll
      for (int r = 0; r < 8; ++r) {
        const int brow = (m << 4) + r + 8 * half;
        hat[(((size_t)brow * I_SZ + i) * C_SZ + c) * DOUT + lane16] = acc[r];
      }
    }
  }
}

// ---------------------------------------------------------------------------
// Kernel B: 3 fused routing iterations. One block per batch element,
// 512 threads. hat is streamed from workspace (L2-resident: 134 MB < 192 MB).
//   c   = softmax(b, axis=C)
//   s   = sum_i c * hat           (thread <-> (c,e), private accumulator)
//   v   = s / sqrt(||s||^2 + eps)
//   b  += sum_e hat * v           (thread <-> (i_local,c), coalesced 64B/thread)
// ---------------------------------------------------------------------------
__global__ __launch_bounds__(512) void caps_routing(
    const float* __restrict__ hat,   // (B, I, C, DOUT)
    const float* __restrict__ b0,    // (1, I, C, 1) -> (I, C)
    float* __restrict__ bbuf_g,      // (B, I, C) scratch
    float* __restrict__ out)         // (B, C, DOUT)
{
  const int bb = blockIdx.x;
  const int t  = threadIdx.x;
  const int c_ce = t >> 4, e_ce = t & 15;   // (c, e) mapping (512 = 32*16)
  const int il_ic = t >> 5, c_ic = t & 31;  // (i_local, c) mapping (512 = 16*32)

  __shared__ float psh[16 * 32];     // per-chunk logits -> softmax probs
  __shared__ float ssh[C_SZ * DOUT];
  __shared__ float vsh[C_SZ * DOUT];

  const float* hatb = hat + (size_t)bb * I_SZ * C_SZ * DOUT;
  float* bbuf = bbuf_g + (size_t)bb * I_SZ * C_SZ;

  for (int r = 0; r < 3; ++r) {
    float s_acc = 0.f;

    for (int chunk = 0; chunk < I_SZ / 16; ++chunk) {
      const int i = chunk * 16 + il_ic;
      psh[il_ic * 32 + c_ic] =
          (r == 0) ? b0[i * C_SZ + c_ic] : bbuf[(size_t)i * C_SZ + c_ic];
      __syncthreads();

      if (t < 16) {  // row-wise softmax over C=32 (max-subtracted)
        float m = -3.4e38f;
        for (int c = 0; c < C_SZ; ++c) m = fmaxf(m, psh[t * 32 + c]);
        float sum = 0.f;
        for (int c = 0; c < C_SZ; ++c) {
          float ex = expf(psh[t * 32 + c] - m);
          psh[t * 32 + c] = ex;
          sum += ex;
        }
        const float inv = 1.f / sum;
        for (int c = 0; c < C_SZ; ++c) psh[t * 32 + c] *= inv;
      }
      __syncthreads();

      const float* hp =
          hatb + ((size_t)(chunk * 16) * C_SZ + c_ce) * DOUT + e_ce;
#pragma unroll 4
      for (int il = 0; il < 16; ++il) {
        s_acc += psh[il * 32 + c_ce] * hp[(size_t)il * C_SZ * DOUT];
      }
      __syncthreads();
    }

    // squash over capsule dim
    ssh[t] = s_acc;  // t == c_ce*DOUT + e_ce
    __syncthreads();
    float nrm = 0.f;
#pragma unroll
    for (int e = 0; e < DOUT; ++e) {
      const float x = ssh[c_ce * DOUT + e];
      nrm += x * x;
    }
    const float v = s_acc / sqrtf(nrm + 1e-7f);
    vsh[t] = v;
    if (r == 2) out[(size_t)bb * C_SZ * DOUT + t] = v;
    __syncthreads();

    if (r < 2) {  // agreement update: b += <hat, v>
      for (int chunk = 0; chunk < I_SZ / 16; ++chunk) {
        const int i = chunk * 16 + il_ic;
        const float* hp = hatb + ((size_t)i * C_SZ + c_ic) * DOUT;
        float dot = 0.f;
#pragma unroll
        for (int e = 0; e < DOUT; ++e) dot += hp[e] * vsh[c_ic * DOUT + e];
        const float bold =
            (r == 0) ? b0[i * C_SZ + c_ic] : bbuf[(size_t)i * C_SZ + c_ic];
        bbuf[(size_t)i * C_SZ + c_ic] = bold + dot;
      }
      __syncthreads();  // make bbuf writes visible to next iteration
    }
  }
}

extern "C" void kernel_launch(void* const* d_in, const int* in_sizes, int n_in,
                              void* d_out, int out_size, void* d_ws, size_t ws_size,
                              hipStream_t stream) {
  (void)in_sizes; (void)n_in; (void)out_size; (void)ws_size;
  const float* in = (const float*)d_in[0];  // (64, 1024, 16)
  const float* W  = (const float*)d_in[1];  // (1024, 32, 16, 16)
  const float* b0 = (const float*)d_in[2];  // (1, 1024, 32, 1)
  float* out = (float*)d_out;               // (64, 32, 16)

  float* hat  = (float*)d_ws;                               // 134,217,728 B
  float* bbuf = hat + (size_t)B_SZ * I_SZ * C_SZ * DOUT;    //   8,388,608 B

  caps_hat_wmma<<<I_SZ, 256, 0, stream>>>(in, W, hat);
  caps_routing<<<B_SZ, 512, 0, stream>>>(hat, b0, bbuf, out);
}